// GaussianSplatting_1709396984135
// MI455X (gfx1250) — compile-verified
//
#include <hip/hip_runtime.h>
#include <hip/hip_bf16.h>

// ---- problem constants (match reference) ------------------------------------
static constexpr int   IMG_H = 720;
static constexpr int   IMG_W = 1280;
static constexpr float K_FX  = 800.0f;
static constexpr float K_FY  = 800.0f;
static constexpr float K_CX  = 640.0f;
static constexpr float K_CY  = 360.0f;

// ---- pass A: init winner buffer to -1 (empty pixel) -------------------------
__global__ __launch_bounds__(256) void init_winner_kernel(int* __restrict__ winner, int npix) {
    int i = blockIdx.x * blockDim.x + threadIdx.x;
    if (i < npix) winner[i] = -1;
}

// ---- pass B: scatter argmax of point index per pixel ------------------------
// Processes 4 points per thread: 3x float4 loads -> global_load_b128, fully
// coalesced. atomicMax is non-returning -> global_atomic_max_i32 resolved in L2
// (winner buffer is 3.7 MB, L2-resident on the 192 MB L2).
__global__ __launch_bounds__(256) void splat_max_kernel(
    const float* __restrict__ pos, int n4, int* __restrict__ winner) {
    const float4* __restrict__ p4 = reinterpret_cast<const float4*>(pos);
    const int stride = blockDim.x * gridDim.x;
    for (int t = blockIdx.x * blockDim.x + threadIdx.x; t < n4; t += stride) {
        // Speculative prefetch of next grid-stride iteration's cache lines
        // (gfx1250 global_prefetch_b8; silently dropped if past the array).
        __builtin_prefetch(&p4[3 * (size_t)(t + stride)], 0, 0);

        float4 a = p4[3 * (size_t)t + 0];
        float4 b = p4[3 * (size_t)t + 1];
        float4 c = p4[3 * (size_t)t + 2];
        // AoS unpack: points are (x,y,z) packed; 12 floats = 4 points.
        float px[4] = {a.x, a.w, b.z, c.y};
        float py[4] = {a.y, b.x, b.w, c.z};
        float pz[4] = {a.z, b.y, c.x, c.w};
        const int base = 4 * t;
#pragma unroll
        for (int k = 0; k < 4; ++k) {
            float z = pz[k] + 3.0f;
            float u = K_FX * (px[k] / z) + K_CX;
            float v = K_FY * (py[k] / z) + K_CY;
            // NaN (z==0) fails all compares -> invalid, same as reference.
            if (u >= 0.0f && u < (float)IMG_W && v >= 0.0f && v < (float)IMG_H) {
                int xi = (int)floorf(u);   // u in [0,W)
                int yi = (int)floorf(v);   // v in [0,H)
                atomicMax(&winner[yi * IMG_W + xi], base + k);
            }
        }
    }
}

// Scalar tail for n not divisible by 4 (not hit for N=4M, kept for generality).
__global__ void splat_tail_kernel(const float* __restrict__ pos, int start, int n,
                                  int* __restrict__ winner) {
    int i = start + blockIdx.x * blockDim.x + threadIdx.x;
    if (i >= n) return;
    float x = pos[3 * (size_t)i + 0];
    float y = pos[3 * (size_t)i + 1];
    float z = pos[3 * (size_t)i + 2] + 3.0f;
    float u = K_FX * (x / z) + K_CX;
    float v = K_FY * (y / z) + K_CY;
    if (u >= 0.0f && u < (float)IMG_W && v >= 0.0f && v < (float)IMG_H) {
        int xi = (int)floorf(u);
        int yi = (int)floorf(v);
        atomicMax(&winner[yi * IMG_W + xi], i);
    }
}

// ---- pass C: per-pixel resolve ----------------------------------------------
// Recomputes u,v,weight for the winning point exactly as the reference does.
// Gathers are sparse and hit the tiny splat region (L2-resident).
__global__ __launch_bounds__(256) void resolve_kernel(
    const int* __restrict__ winner,
    const float* __restrict__ pos,
    const float* __restrict__ scl,
    const float* __restrict__ col,
    float* __restrict__ out, int npix) {
    int p = blockIdx.x * blockDim.x + threadIdx.x;
    if (p >= npix) return;
    int w = winner[p];
    float r = 0.0f, g = 0.0f, b = 0.0f;
    if (w >= 0) {
        float x = pos[3 * (size_t)w + 0];
        float y = pos[3 * (size_t)w + 1];
        float z = pos[3 * (size_t)w + 2] + 3.0f;
        float u = K_FX * (x / z) + K_CX;
        float v = K_FY * (y / z) + K_CY;
        float dx = u - floorf(u);
        float dy = v - floorf(v);
        float rad = scl[3 * (size_t)w + 0] * 100.0f;
        float wt = expf(-(dx * dx + dy * dy) / (2.0f * rad * rad));
        r = col[3 * (size_t)w + 0] * wt;
        g = col[3 * (size_t)w + 1] * wt;
        b = col[3 * (size_t)w + 2] * wt;
    }
    // Contiguous 12B per thread across consecutive lanes -> coalesced b96.
    out[3 * (size_t)p + 0] = r;
    out[3 * (size_t)p + 1] = g;
    out[3 * (size_t)p + 2] = b;
}

extern "C" void kernel_launch(void* const* d_in, const int* in_sizes, int n_in,
                              void* d_out, int out_size, void* d_ws, size_t ws_size,
                              hipStream_t stream) {
    // Input order per setup_inputs(): camera_poses (unused), positions, scales, colors.
    const float* positions = (const float*)d_in[1];
    const float* scales    = (const float*)d_in[2];
    const float* colors    = (const float*)d_in[3];
    float*       out       = (float*)d_out;

    const int n    = in_sizes[1] / 3;       // number of points
    const int npix = IMG_W * IMG_H;         // 921600
    int* winner = (int*)d_ws;               // npix * 4 bytes scratch

    // Pass A: winner = -1 everywhere.
    init_winner_kernel<<<(npix + 255) / 256, 256, 0, stream>>>(winner, npix);

    // Pass B: scatter argmax. 4 points/thread, grid-stride with ~4 iters/thread.
    const int n4 = n / 4;
    if (n4 > 0) {
        int blocks = 1024;
        int needed = (n4 + 255) / 256;
        if (needed < blocks) blocks = needed;
        splat_max_kernel<<<blocks, 256, 0, stream>>>(positions, n4, winner);
    }
    const int tail_start = n4 * 4;
    if (tail_start < n) {
        int tail = n - tail_start;
        splat_tail_kernel<<<(tail + 63) / 64, 64, 0, stream>>>(positions, tail_start, n, winner);
    }

    // Pass C: resolve winners into the image.
    resolve_kernel<<<(npix + 255) / 256, 256, 0, stream>>>(winner, positions, scales, colors, out, npix);
}